// DGLFeatureGAT_23922967839177
// MI455X (gfx1250) — compile-verified
//
#include <hip/hip_runtime.h>

typedef __attribute__((ext_vector_type(2))) float v2f;
typedef __attribute__((ext_vector_type(8))) float v8f;

#define BATCH 256
#define WDIM 256
#define FDIM 256
#define HEADS 4
#define DDIM 8
#define HD 32
#define NEG_SLOPE 0.2f

// LDS layout (floats). Region 0 (fc_w) is dead after phase 1 and is reused
// as the attention output buffer in phase 3.
#define OFF_FCW    0        // [32][256]  fc_w, 32KB   -> reused as attn_out [256][32]
#define OFF_PROJW  8192     // [256][32]  proj_w, 32KB
#define OFF_FEAT   16384    // [256][32]  feat, 32KB
#define OFF_EL     24576    // [4][256]
#define OFF_ER     25600    // [4][256]
#define OFF_ATTL   26624    // [32]
#define OFF_ATTR   26656    // [32]
#define OFF_BIAS   26688    // [32]
#define OFF_PROJB  26720    // [256]
#define SMEM_FLOATS 26976   // ~105 KB of the 320 KB WGP LDS

__global__ __launch_bounds__(256)
void gat_fused_kernel(const float* __restrict__ x,
                      const float* __restrict__ fc_w,
                      const float* __restrict__ attn_l,
                      const float* __restrict__ attn_r,
                      const float* __restrict__ gat_bias,
                      const float* __restrict__ proj_w,
                      const float* __restrict__ proj_b,
                      float* __restrict__ out) {
    __shared__ float smem[SMEM_FLOATS];
    float* s_fcw   = smem + OFF_FCW;
    float* s_projw = smem + OFF_PROJW;
    float* s_feat  = smem + OFF_FEAT;
    float* s_el    = smem + OFF_EL;
    float* s_er    = smem + OFF_ER;
    float* s_attl  = smem + OFF_ATTL;
    float* s_attr  = smem + OFF_ATTR;
    float* s_bias  = smem + OFF_BIAS;
    float* s_projb = smem + OFF_PROJB;

    const int t    = threadIdx.x;
    const int b    = blockIdx.x;
    const int lane = t & 31;          // wave32
    const int wv   = t >> 5;          // 8 waves per block
    const int lh   = lane >> 4;       // half-wave select (K split for f32 WMMA)
    const int ll   = lane & 15;       // row/col within fragment

    const float* xb   = x   + (size_t)b * (WDIM * FDIM);
    float*       outb = out + (size_t)b * (WDIM * FDIM);

    // ---------------- phase 0: stage weights into LDS ----------------
    for (int idx = t; idx < 8192; idx += 256) {
        s_fcw[idx]   = fc_w[idx];     // fc_w[o][w], row length 256
        s_projw[idx] = proj_w[idx];   // proj_w[wo][o], row length 32
    }
    if (t < 32) {
        s_attl[t] = attn_l[t];
        s_attr[t] = attn_r[t];
        s_bias[t] = gat_bias[t];
    }
    s_projb[t] = proj_b[t];
    __syncthreads();

    // ---------------- phase 1: feat = x[b]^T @ fc_w^T  (M=256,K=256,N=32) ----
    // wave wv owns feat rows [32*wv, 32*wv+32): 2 M-tiles x 2 N-tiles.
    {
        const int m0 = wv * 32;
        v8f acc[2][2] = {};
        for (int k = 0; k < WDIM; k += 4) {
            const int w0 = k + 2 * lh;            // K pair for this half-wave
            if ((k & 31) == 0 && k + 32 < WDIM) { // prefetch next K slab of x
                __builtin_prefetch(xb + (size_t)(k + 32) * FDIM + m0 + ll, 0, 1);
            }
            v2f a[2];
#pragma unroll
            for (int mt = 0; mt < 2; ++mt) {
                const int f = m0 + mt * 16 + ll;  // A row (lane-contiguous -> coalesced)
                a[mt].x = xb[(size_t)w0 * FDIM + f];
                a[mt].y = xb[(size_t)(w0 + 1) * FDIM + f];
            }
            v2f bm[2];
#pragma unroll
            for (int nt = 0; nt < 2; ++nt) {
                const int o = nt * 16 + ll;       // B col; w0 even -> 8B aligned b64
                bm[nt] = *reinterpret_cast<const v2f*>(&s_fcw[o * WDIM + w0]);
            }
#pragma unroll
            for (int mt = 0; mt < 2; ++mt)
#pragma unroll
                for (int nt = 0; nt < 2; ++nt)
                    acc[mt][nt] = __builtin_amdgcn_wmma_f32_16x16x4_f32(
                        false, a[mt], false, bm[nt], (short)0, acc[mt][nt],
                        false, false);
        }
        // D layout: VGPR r of lane -> M = 8*lh + r, N = ll
#pragma unroll
        for (int mt = 0; mt < 2; ++mt)
#pragma unroll
            for (int nt = 0; nt < 2; ++nt)
#pragma unroll
                for (int r = 0; r < 8; ++r) {
                    const int f = m0 + mt * 16 + lh * 8 + r;
                    const int o = nt * 16 + ll;
                    s_feat[f * HD + o] = acc[mt][nt][r];
                }
    }
    __syncthreads();

    // ---------------- phase 2: el/er = feat . attn_{l,r} ----------------
    {
        const int f = t;
#pragma unroll
        for (int h = 0; h < HEADS; ++h) {
            float sl = 0.f, sr = 0.f;
#pragma unroll
            for (int d = 0; d < DDIM; ++d) {
                const float v = s_feat[f * HD + h * DDIM + d];
                sl += v * s_attl[h * DDIM + d];
                sr += v * s_attr[h * DDIM + d];
            }
            s_el[h * FDIM + f] = sl;   // [h][src] for broadcast-friendly reads
            s_er[h * FDIM + f] = sr;
        }
    }
    __syncthreads();

    // ---------------- phase 3: fused edge-softmax + aggregation ----------------
    // alpha[b,i,s,h] = softmax_s(leaky(el[s]+er[i])); out[i,h,:] = sum_s alpha*feat[s,h,:]
    float* s_attnout = s_fcw;  // fc_w region is dead; reuse as [256][32]
    for (int rep = 0; rep < 4; ++rep) {
        const int idx = rep * 256 + t;
        const int i = idx >> 2;           // dst node
        const int h = idx & 3;            // head
        const float eri = s_er[h * FDIM + i];
        const float* elh = &s_el[h * FDIM];
        float m = -3.402823466e38f;
        for (int s = 0; s < FDIM; ++s) {
            float e = elh[s] + eri;
            e = e > 0.f ? e : NEG_SLOPE * e;
            m = fmaxf(m, e);
        }
        float denom = 0.f;
        float accd[DDIM] = {};
        for (int s = 0; s < FDIM; ++s) {
            float e = elh[s] + eri;
            e = e > 0.f ? e : NEG_SLOPE * e;
            const float w = __expf(e - m);
            denom += w;
            const float* fs = &s_feat[s * HD + h * DDIM];
#pragma unroll
            for (int d = 0; d < DDIM; ++d) accd[d] += w * fs[d];
        }
        const float inv = 1.f / denom;
#pragma unroll
        for (int d = 0; d < DDIM; ++d)
            s_attnout[i * HD + h * DDIM + d] =
                accd[d] * inv + s_bias[h * DDIM + d];
    }
    __syncthreads();

    // ---------------- phase 4: out[b,wo,f] = proj_w[wo,:] . attn_out[f,:] + proj_b[wo]
    // M = wo (so stores are lane-contiguous in f -> coalesced), N = f, K = 32.
    {
        const int m0 = wv * 32;  // this wave's wo rows
        for (int nt = 0; nt < 16; ++nt) {
            const int n0 = nt * 16;
#pragma unroll
            for (int mt = 0; mt < 2; ++mt) {
                const int mb = m0 + mt * 16;
                v8f acc = {};
#pragma unroll
                for (int k = 0; k < HD; k += 4) {
                    const int o = k + 2 * lh;   // even -> 8B aligned b64 loads
                    v2f a  = *reinterpret_cast<const v2f*>(&s_projw[(mb + ll) * HD + o]);
                    v2f bm = *reinterpret_cast<const v2f*>(&s_attnout[(n0 + ll) * HD + o]);
                    acc = __builtin_amdgcn_wmma_f32_16x16x4_f32(
                        false, a, false, bm, (short)0, acc, false, false);
                }
#pragma unroll
                for (int r = 0; r < 8; ++r) {
                    const int wo = mb + lh * 8 + r;
                    const int f  = n0 + ll;     // 16 consecutive floats per half-wave
                    outb[(size_t)wo * FDIM + f] = acc[r] + s_projb[wo];
                }
            }
        }
    }
}

extern "C" void kernel_launch(void* const* d_in, const int* in_sizes, int n_in,
                              void* d_out, int out_size, void* d_ws, size_t ws_size,
                              hipStream_t stream) {
    const float* x        = (const float*)d_in[0];
    const float* fc_w     = (const float*)d_in[1];
    const float* attn_l   = (const float*)d_in[2];
    const float* attn_r   = (const float*)d_in[3];
    const float* gat_bias = (const float*)d_in[4];
    const float* proj_w   = (const float*)d_in[5];
    const float* proj_b   = (const float*)d_in[6];
    float* out = (float*)d_out;
    (void)in_sizes; (void)n_in; (void)out_size; (void)d_ws; (void)ws_size;

    gat_fused_kernel<<<dim3(BATCH), dim3(256), 0, stream>>>(
        x, fc_w, attn_l, attn_r, gat_bias, proj_w, proj_b, out);
}